// NFPN_RPS_Net_72438918414867
// MI455X (gfx1250) — compile-verified
//
#include <hip/hip_runtime.h>

typedef float v2f __attribute__((ext_vector_type(2)));
typedef float v8f __attribute__((ext_vector_type(8)));
typedef int   v2i __attribute__((ext_vector_type(2)));

#define NFPN_EPS2 1e-10f   // EPS=1e-5 on the 6-vector L2 norm -> compare squared
#define NFPN_MAX_DEPTH 100

__global__ __launch_bounds__(256) void nfpn_rps_kernel(
    const float* __restrict__ din,
    const float* __restrict__ W1,
    const float* __restrict__ b1,
    const float* __restrict__ W2,
    const float* __restrict__ b2,
    float* __restrict__ out,
    int B)
{
    const int gid  = blockIdx.x * blockDim.x + threadIdx.x;
    const int lane = threadIdx.x & 31;
    const bool low = lane < 16;

    int s = gid;
    const bool active = (s < B);
    if (!active) s = B - 1;           // clamp: keep EXEC all-1s for WMMA

    // ---- load sample d[s] ----
    const float d0 = din[3*s + 0];
    const float d1 = din[3*s + 1];
    const float d2 = din[3*s + 2];

    // ---- Qd = leaky_relu(d @ W1^T + b1) (weights uniform -> scalar loads) ----
    float q[6];
#pragma unroll
    for (int j = 0; j < 6; ++j) {
        float t = fmaf(d0, W1[3*j + 0], fmaf(d1, W1[3*j + 1], fmaf(d2, W1[3*j + 2], b1[j])));
        q[j] = (t >= 0.f) ? t : 0.1f * t;
    }

    // ---- constant WMMA A operands: A(16x8) rows 0-5 = -W2, rows 8-13 = -W2 (dup), rest 0
    // A 16x4 f32 layout: lanes 0-15 -> M=lane, K=0 (v.x), K=1 (v.y); lanes 16-31 -> M=lane-16, K=2,3
    const int  M     = lane & 15;
    const int  rr    = M & 7;
    const bool valid = (rr < 6);
    v2f a1, a2;
    a1.x = valid ? -W2[rr*6 + (low ? 0 : 2)] : 0.f;
    a1.y = valid ? -W2[rr*6 + (low ? 1 : 3)] : 0.f;
    a2.x = (valid && low) ? -W2[rr*6 + 4] : 0.f;   // high-lane K cols 6,7 are padding -> 0
    a2.y = (valid && low) ? -W2[rr*6 + 5] : 0.f;

    // C = -b2 broadcast; covers rows r (low lanes) and r+8 (high lanes) identically
    v8f cinit;
#pragma unroll
    for (int j = 0; j < 6; ++j) cinit[j] = -b2[j];
    cinit[6] = 0.f; cinit[7] = 0.f;

    float u[6];
#pragma unroll
    for (int j = 0; j < 6; ++j) u[j] = 1.0f / 3.0f;

    // Exchange wave halves of (a,b) in one VALU op when v_permlane16_swap is available:
    //   result.x = { a.lanes0-15 | b.lanes0-15 shifted to 16-31 }   (== low ? a : b-from-partner)
    //   result.y = { a.lanes16-31 shifted to 0-15 | b.lanes16-31 }  (== low ? a-from-partner : b)
    auto swap_halves = [&](float a, float b) -> v2f {
        v2f o;
#if __has_builtin(__builtin_amdgcn_permlane16_swap)
        v2i r = __builtin_amdgcn_permlane16_swap(__float_as_int(a), __float_as_int(b),
                                                 false, false);
        o.x = __int_as_float(r.x);
        o.y = __int_as_float(r.y);
#else
        float as = __shfl_xor(a, 16, 32);
        float bs = __shfl_xor(b, 16, 32);
        o.x = low ? a : bs;
        o.y = low ? as : b;
#endif
        return o;
    };

    // y = -((u+q) @ W2^T + b2) for this lane's sample; 4 f32 WMMAs cover 32 samples/wave
    auto matvec = [&](const float* x, float* y) {
        v2f B1, B2, B1p, B2p;
        // set0 B (samples in lanes 0-15): low lanes = K rows 0/1, high lanes = K rows 2/3
        // set1 B (samples in lanes 16-31): low lanes = partner's K rows 0/1, high = K rows 2/3
        v2f s02 = swap_halves(x[0], x[2]);  B1.x = s02.x;  B1p.x = s02.y;
        v2f s13 = swap_halves(x[1], x[3]);  B1.y = s13.x;  B1p.y = s13.y;
        v2f s4z = swap_halves(x[4], 0.f);   B2.x = s4z.x;  B2p.x = s4z.y;  // K rows 6/7 pad -> 0
        v2f s5z = swap_halves(x[5], 0.f);   B2.y = s5z.x;  B2p.y = s5z.y;
        v8f acc = cinit;
        acc  = __builtin_amdgcn_wmma_f32_16x16x4_f32(false, a1, false, B1,  (short)0, acc,  false, false);
        acc  = __builtin_amdgcn_wmma_f32_16x16x4_f32(false, a2, false, B2,  (short)0, acc,  false, false);
        v8f accp = cinit;
        accp = __builtin_amdgcn_wmma_f32_16x16x4_f32(false, a1, false, B1p, (short)0, accp, false, false);
        accp = __builtin_amdgcn_wmma_f32_16x16x4_f32(false, a2, false, B2p, (short)0, accp, false, false);
        // D row r (low lanes) == row r+8 (high lanes) thanks to duplicated A rows
#pragma unroll
        for (int j = 0; j < 6; ++j) y[j] = low ? acc[j] : accp[j];
    };

    // Euclidean projection onto the 3-simplex (closed form for a=3)
    auto proj3 = [&](float a, float b, float c, float& o0, float& o1, float& o2) {
        float hi  = fmaxf(fmaxf(a, b), c);
        float lo  = fminf(fminf(a, b), c);
        float sum = a + b + c;
        float mid = sum - hi - lo;
        float c2  = hi + mid;
        bool cond2 = (mid - hi + 1.0f) > 0.0f;        // u2 - (c2-1)/2 > 0
        bool cond3 = (3.0f*lo - sum + 1.0f) > 0.0f;   // u3 - (c3-1)/3 > 0
        int rho = 1 + (cond2 ? 1 : 0) + (cond3 ? 1 : 0);
        float cs    = (rho == 3) ? sum : ((rho == 2) ? c2 : hi);
        float rinv  = (rho == 3) ? (1.0f/3.0f) : ((rho == 2) ? 0.5f : 1.0f);
        float theta = (cs - 1.0f) * rinv;
        o0 = fmaxf(a - theta, 0.f);
        o1 = fmaxf(b - theta, 0.f);
        o2 = fmaxf(c - theta, 0.f);
    };

    // ---- fixed-point iteration (wave-uniform loop; EXEC stays all-1s) ----
    int depth = 0;
    for (;;) {
        float x[6], y[6], un[6];
#pragma unroll
        for (int j = 0; j < 6; ++j) x[j] = u[j] + q[j];
        matvec(x, y);
        proj3(y[0], y[1], y[2], un[0], un[1], un[2]);
        proj3(y[3], y[4], y[5], un[3], un[4], un[5]);
        float r2 = 0.f;
#pragma unroll
        for (int j = 0; j < 6; ++j) { float dd = un[j] - u[j]; r2 = fmaf(dd, dd, r2); u[j] = un[j]; }
        ++depth;
        if (depth >= NFPN_MAX_DEPTH) break;
        if (__ballot(r2 > NFPN_EPS2) == 0ULL) break;  // all lanes in wave converged
    }

    // ---- one final attached step ----
    float x[6], y[6], o[6];
#pragma unroll
    for (int j = 0; j < 6; ++j) x[j] = u[j] + q[j];
    matvec(x, y);
    proj3(y[0], y[1], y[2], o[0], o[1], o[2]);
    proj3(y[3], y[4], y[5], o[3], o[4], o[5]);

    if (active) {
        float2* op = reinterpret_cast<float2*>(out + 6l * s);  // 24B/sample -> 8B aligned
        op[0] = make_float2(o[0], o[1]);
        op[1] = make_float2(o[2], o[3]);
        op[2] = make_float2(o[4], o[5]);
    }
}

extern "C" void kernel_launch(void* const* d_in, const int* in_sizes, int n_in,
                              void* d_out, int out_size, void* d_ws, size_t ws_size,
                              hipStream_t stream) {
    const float* d  = (const float*)d_in[0];
    const float* W1 = (const float*)d_in[1];
    const float* b1 = (const float*)d_in[2];
    const float* W2 = (const float*)d_in[3];
    const float* b2 = (const float*)d_in[4];
    float* out = (float*)d_out;

    const int B = in_sizes[0] / 3;          // d is [B,3]
    const int threads = 256;                 // 8 wave32 per block
    const int blocks  = (B + threads - 1) / threads;
    hipLaunchKernelGGL(nfpn_rps_kernel, dim3(blocks), dim3(threads), 0, stream,
                       d, W1, b1, W2, b2, out, B);
}